// _Expert_30734785970328
// MI455X (gfx1250) — compile-verified
//
#include <hip/hip_runtime.h>
#include <math.h>

typedef float v2f __attribute__((ext_vector_type(2)));
typedef float v8f __attribute__((ext_vector_type(8)));
typedef int   vi4 __attribute__((vector_size(4 * sizeof(int))));  // matches builtin param type

#define AS1 __attribute__((address_space(1)))
#define AS3 __attribute__((address_space(3)))

#if defined(__gfx1250__) && __has_builtin(__builtin_amdgcn_global_load_async_to_lds_b128) && \
    __has_builtin(__builtin_amdgcn_s_wait_asynccnt)
#define USE_ASYNC 1
#else
#define USE_ASYNC 0
#endif

static constexpr int NEXP      = 8;
static constexpr int D_MODEL   = 2048;
static constexpr int D_HIDDEN  = 8192;
static constexpr int TPE       = 2048;   // tokens per expert

// Tiling
static constexpr int BM = 128;
static constexpr int BN = 128;
static constexpr int BK = 32;
static constexpr int LDSS = BK + 4;      // 36-dword row stride: conflict-free + 16B aligned

// C[M,N] = A[M,K] * B[N,K]^T + bias[N], optional tanh-gelu.
// Both operands are row-major with K contiguous (true for X/W1 and H/W2).
template <bool DO_GELU>
__global__ __launch_bounds__(256) void moe_gemm_f32_wmma(
    const float* __restrict__ A,
    const float* __restrict__ B,
    const float* __restrict__ bias,
    float* __restrict__ C,
    int M, int N, int K)
{
    __shared__ float lA[2][BM * LDSS];
    __shared__ float lB[2][BN * LDSS];

    const int tid  = threadIdx.x;
    const int wave = tid >> 5;
    const int lane = tid & 31;
    const int half = lane >> 4;   // 0: k,k+1 / C rows 0..7 ; 1: k+2,k+3 / C rows 8..15
    const int l16  = lane & 15;

    const int tile_m = blockIdx.y * BM;
    const int tile_n = blockIdx.x * BN;

    // 8 waves -> 2 (M) x 4 (N), each wave computes 64 x 32
    const int wm = (wave & 1) * 64;
    const int wn = (wave >> 1) * 32;

    v8f acc[4][2];
#pragma unroll
    for (int i = 0; i < 4; ++i)
#pragma unroll
        for (int j = 0; j < 2; ++j)
#pragma unroll
            for (int r = 0; r < 8; ++r) acc[i][j][r] = 0.0f;

    // global->LDS staging: 256 threads, each moves 4+4 float4 per K-slab
    const int lrow = tid >> 3;         // 0..31
    const int lcol = (tid & 7) << 2;   // 0,4,...,28

    const float* gA = A + (size_t)(tile_m + lrow) * K + lcol;
    const float* gB = B + (size_t)(tile_n + lrow) * K + lcol;
    const int nslab = K / BK;

#if USE_ASYNC
    auto issue = [&](int buf, int k0) {
#pragma unroll
        for (int r = 0; r < BM; r += 32)
            __builtin_amdgcn_global_load_async_to_lds_b128(
                (AS1 vi4*)(uintptr_t)(gA + (size_t)r * K + k0),
                (AS3 vi4*)(uintptr_t)&lA[buf][(r + lrow) * LDSS + lcol], 0, 0);
#pragma unroll
        for (int r = 0; r < BN; r += 32)
            __builtin_amdgcn_global_load_async_to_lds_b128(
                (AS1 vi4*)(uintptr_t)(gB + (size_t)r * K + k0),
                (AS3 vi4*)(uintptr_t)&lB[buf][(r + lrow) * LDSS + lcol], 0, 0);
    };
    issue(0, 0);
#else
    float4 ra[4], rb[4];
    auto gload = [&](int k0) {
#pragma unroll
        for (int r = 0; r < 4; ++r) ra[r] = *(const float4*)(gA + (size_t)(r * 32) * K + k0);
#pragma unroll
        for (int r = 0; r < 4; ++r) rb[r] = *(const float4*)(gB + (size_t)(r * 32) * K + k0);
    };
    auto sstore = [&](int buf) {
#pragma unroll
        for (int r = 0; r < 4; ++r) *(float4*)&lA[buf][(r * 32 + lrow) * LDSS + lcol] = ra[r];
#pragma unroll
        for (int r = 0; r < 4; ++r) *(float4*)&lB[buf][(r * 32 + lrow) * LDSS + lcol] = rb[r];
    };
    gload(0);
    sstore(0);
#endif

    for (int s = 0; s < nslab; ++s) {
        const int cur = s & 1;
#if USE_ASYNC
        __builtin_amdgcn_s_wait_asynccnt(0);   // this wave's async writes landed
        __syncthreads();                       // everyone's landed; prev compute done
        if (s + 1 < nslab) issue(cur ^ 1, (s + 1) * BK);
#else
        __syncthreads();                       // LDS stores from prev iter visible
        if (s + 1 < nslab) gload((s + 1) * BK);  // global loads in flight during compute
#endif

#pragma unroll
        for (int kk = 0; kk < BK; kk += 4) {
            v2f a[4], b[2];
#pragma unroll
            for (int i = 0; i < 4; ++i)
                a[i] = *(const v2f*)&lA[cur][(wm + i * 16 + l16) * LDSS + kk + 2 * half];
#pragma unroll
            for (int j = 0; j < 2; ++j)
                b[j] = *(const v2f*)&lB[cur][(wn + j * 16 + l16) * LDSS + kk + 2 * half];
#pragma unroll
            for (int i = 0; i < 4; ++i)
#pragma unroll
                for (int j = 0; j < 2; ++j)
                    acc[i][j] = __builtin_amdgcn_wmma_f32_16x16x4_f32(
                        false, a[i], false, b[j], (short)0, acc[i][j], false, false);
        }

#if !USE_ASYNC
        if (s + 1 < nslab) sstore(cur ^ 1);    // buf last read before this iter's barrier
#endif
    }

    // Epilogue: C/D layout -> lanes 0-15: M=r, N=l16 ; lanes 16-31: M=8+r, N=l16
#pragma unroll
    for (int i = 0; i < 4; ++i) {
        const int row_base = tile_m + wm + i * 16 + half * 8;
#pragma unroll
        for (int j = 0; j < 2; ++j) {
            const int col = tile_n + wn + j * 16 + l16;
            const float bv = bias[col];
#pragma unroll
            for (int r = 0; r < 8; ++r) {
                float v = acc[i][j][r] + bv;
                if (DO_GELU) {
                    // jax.nn.gelu default (approximate=True, tanh form)
                    const float u = 0.7978845608028654f * (v + 0.044715f * v * v * v);
                    v = 0.5f * v * (1.0f + tanhf(u));
                }
                C[(size_t)(row_base + r) * N + col] = v;
            }
        }
    }
}

extern "C" void kernel_launch(void* const* d_in, const int* in_sizes, int n_in,
                              void* d_out, int out_size, void* d_ws, size_t ws_size,
                              hipStream_t stream) {
    const float* inp = (const float*)d_in[0];  // [E*TPE, D_MODEL]
    const float* w1  = (const float*)d_in[1];  // [E, D_HIDDEN, D_MODEL]
    const float* b1  = (const float*)d_in[2];  // [E, D_HIDDEN]
    const float* w2  = (const float*)d_in[3];  // [E, D_MODEL, D_HIDDEN]
    const float* b2  = (const float*)d_in[4];  // [E, D_MODEL]
    // d_in[5] fwd_expert_count: equal counts, layout is static -> unused.

    float* out = (float*)d_out;                // [E*TPE, D_MODEL]
    float* H   = (float*)d_ws;                 // per-expert hidden buffer, TPE*D_HIDDEN f32 = 64 MiB

    const dim3 blk(256);
    const dim3 g1(D_HIDDEN / BN, TPE / BM);    // 64 x 16 workgroups
    const dim3 g2(D_MODEL  / BN, TPE / BM);    // 16 x 16 workgroups

    // Per-expert pipelining: H (64 MiB) + w2_e (64 MiB) fit in the 192 MB L2,
    // so the hidden activation largely avoids HBM round-trips.
    for (int e = 0; e < NEXP; ++e) {
        const float* xe  = inp + (size_t)e * TPE * D_MODEL;
        const float* w1e = w1  + (size_t)e * D_HIDDEN * D_MODEL;
        const float* b1e = b1  + (size_t)e * D_HIDDEN;
        const float* w2e = w2  + (size_t)e * D_MODEL * D_HIDDEN;
        const float* b2e = b2  + (size_t)e * D_MODEL;
        float*       ye  = out + (size_t)e * TPE * D_MODEL;

        moe_gemm_f32_wmma<true><<<g1, blk, 0, stream>>>(xe, w1e, b1e, H, TPE, D_HIDDEN, D_MODEL);
        moe_gemm_f32_wmma<false><<<g2, blk, 0, stream>>>(H, w2e, b2e, ye, TPE, D_MODEL, D_HIDDEN);
    }
}